// myGNN_5257039970767
// MI455X (gfx1250) — compile-verified
//
#include <hip/hip_runtime.h>
#include <math.h>

#define N_NODES 50000
#define N_EDGES 800000
#define F_IN    128
#define HID     96
#define N_LAYERS 4

typedef __attribute__((ext_vector_type(2))) float v2f;
typedef __attribute__((ext_vector_type(8))) float v8f;

// ---------------------------------------------------------------- utilities

__device__ __forceinline__ float wave_sum32(float v) {
#pragma unroll
    for (int off = 16; off > 0; off >>= 1) v += __shfl_xor(v, off, 32);
    return v;
}

__device__ __forceinline__ float atomicMaxF(float* addr, float val) {
    int* ai = (int*)addr;
    int old = __float_as_int(*addr);
    while (__int_as_float(old) < val) {
        int assumed = old;
        old = atomicCAS(ai, assumed, __float_as_int(val));
        if (old == assumed) break;
    }
    return __int_as_float(old);
}

// ---------------------------------------------------------------- WMMA GEMM
// C[M x Ncol] = A[M x K] @ W[K x Ncol] + bias[Ncol]
// One wave (32 threads) computes one 16x16 tile of C using
// V_WMMA_F32_16X16X4_F32 (fp32 A/B, fp32 accumulate).
// grid = (M/16, Ncol/16), block = 32.
__global__ void k_gemm_bias(const float* __restrict__ A,
                            const float* __restrict__ W,
                            const float* __restrict__ bias,
                            float* __restrict__ C,
                            int M, int K, int Ncol) {
    const int lane = threadIdx.x;          // 0..31
    const int half = lane >> 4;            // 0 or 1
    const int l16  = lane & 15;
    const int tm   = blockIdx.x;
    const int tn   = blockIdx.y;
    const int mr   = (tm << 4) + l16;      // A row this lane reads
    const int nc   = (tn << 4) + l16;      // B/C column this lane owns

    v8f c = {};
    for (int k0 = 0; k0 < K; k0 += 4) {
        const int ka = k0 + (half << 1);
        // A 16x4 fragment: lane holds A[m][ka], A[m][ka+1] (contiguous)
        v2f a;
        const float* ap = A + (size_t)mr * K + ka;
        a.x = ap[0];
        a.y = ap[1];
        // B 4x16 fragment: lane holds W[ka][nc], W[ka+1][nc]
        v2f b;
        b.x = W[(size_t)ka * Ncol + nc];
        b.y = W[(size_t)(ka + 1) * Ncol + nc];
        c = __builtin_amdgcn_wmma_f32_16x16x4_f32(
                /*neg_a=*/false, a, /*neg_b=*/false, b,
                /*c_mod=*/(short)0, c, /*reuse_a=*/false, /*reuse_b=*/false);
    }

    const float bn = bias[nc];
    const int mbase = (tm << 4) + (half << 3);
#pragma unroll
    for (int r = 0; r < 8; ++r) {
        C[(size_t)(mbase + r) * Ncol + nc] = c[r] + bn;
    }
}

// ---------------------------------------------------------------- LayerNorm + ReLU
// One wave per node, lane handles cols {l, l+32, l+64}.
// block = (32, 8), grid = N/8.
__global__ void k_ln_relu(const float* __restrict__ h,
                          const float* __restrict__ g,
                          const float* __restrict__ b,
                          float* __restrict__ out) {
    const int node = blockIdx.x * blockDim.y + threadIdx.y;
    const int lane = threadIdx.x;
    const float* row = h + (size_t)node * HID;

    float x0 = row[lane];
    float x1 = row[lane + 32];
    float x2 = row[lane + 64];

    const float mu  = wave_sum32(x0 + x1 + x2) * (1.0f / HID);
    const float sq  = wave_sum32(x0 * x0 + x1 * x1 + x2 * x2) * (1.0f / HID);
    const float var = sq - mu * mu;
    const float rs  = rsqrtf(var + 1e-5f);

    float* orow = out + (size_t)node * HID;
#pragma unroll
    for (int j = 0; j < 3; ++j) {
        const int k = lane + j * 32;
        const float x = (j == 0) ? x0 : (j == 1) ? x1 : x2;
        const float v = (x - mu) * rs * g[k] + b[k];
        orow[k] = v > 0.0f ? v : 0.0f;
    }
}

// ---------------------------------------------------------------- per-layer init
__global__ void k_init(float* __restrict__ agg,
                       float* __restrict__ segmax,
                       float* __restrict__ segsum) {
    const int i = blockIdx.x * blockDim.x + threadIdx.x;
    if (i < N_NODES * HID) agg[i] = 0.0f;
    if (i < N_NODES) {
        segmax[i] = -INFINITY;
        segsum[i] = 0.0f;
    }
}

// ---------------------------------------------------------------- edge alpha
// alpha[e] = sum_k lrelu(xl[src][k] + xr[dst][k] + ew*We[k], 0.2) * att[k]
// One wave per edge; also atomic segment-max into segmax[dst].
// block = (32, 8), grid = E/8.
__global__ void k_edge_alpha(const float* __restrict__ xl,
                             const float* __restrict__ xr,
                             const long long* __restrict__ src,
                             const long long* __restrict__ dst,
                             const float* __restrict__ ew,
                             const float* __restrict__ We,
                             const float* __restrict__ att,
                             float* __restrict__ alpha,
                             float* __restrict__ segmax) {
    const int e = blockIdx.x * blockDim.y + threadIdx.y;
    const int lane = threadIdx.x;
    const long long s = src[e];
    const long long d = dst[e];
    const float w = ew[e];
    const float* xls = xl + (size_t)s * HID;
    const float* xrd = xr + (size_t)d * HID;

    float acc = 0.0f;
#pragma unroll
    for (int j = 0; j < 3; ++j) {
        const int k = lane + j * 32;
        float v = xls[k] + xrd[k] + w * We[k];
        v = v > 0.0f ? v : 0.2f * v;
        acc += v * att[k];
    }
    const float a = wave_sum32(acc);
    if (lane == 0) {
        alpha[e] = a;
        atomicMaxF(&segmax[d], a);
    }
}

// ---------------------------------------------------------------- softmax exp + segment sum
__global__ void k_edge_exp(const float* __restrict__ alpha,
                           const float* __restrict__ segmax,
                           const long long* __restrict__ dst,
                           float* __restrict__ aexp,
                           float* __restrict__ segsum) {
    const int e = blockIdx.x * blockDim.x + threadIdx.x;
    const long long d = dst[e];
    const float a = expf(alpha[e] - segmax[d]);
    aexp[e] = a;
    atomicAdd(&segsum[d], a);
}

// ---------------------------------------------------------------- weighted scatter-add
// agg[dst] += xl[src] * (aexp[e] / (segsum[dst] + eps)); wave per edge.
__global__ void k_edge_agg(const float* __restrict__ xl,
                           const long long* __restrict__ src,
                           const long long* __restrict__ dst,
                           const float* __restrict__ aexp,
                           const float* __restrict__ segsum,
                           float* __restrict__ agg) {
    const int e = blockIdx.x * blockDim.y + threadIdx.y;
    const int lane = threadIdx.x;
    const long long s = src[e];
    const long long d = dst[e];
    const float w = aexp[e] / (segsum[d] + 1e-16f);
    const float* xls = xl + (size_t)s * HID;
    float* ad = agg + (size_t)d * HID;
#pragma unroll
    for (int j = 0; j < 3; ++j) {
        const int k = lane + j * 32;
        atomicAdd(&ad[k], xls[k] * w);
    }
}

// ---------------------------------------------------------------- residual + bias
__global__ void k_residual(float* __restrict__ h,
                           const float* __restrict__ agg,
                           const float* __restrict__ bias) {
    const int i = blockIdx.x * blockDim.x + threadIdx.x;
    h[i] += agg[i] + bias[i % HID];
}

// ---------------------------------------------------------------- final FC (HID -> 1)
__global__ void k_fc(const float* __restrict__ hn,
                     const float* __restrict__ fcW,
                     const float* __restrict__ fcb,
                     float* __restrict__ out) {
    const int node = blockIdx.x * blockDim.y + threadIdx.y;
    const int lane = threadIdx.x;
    const float* row = hn + (size_t)node * HID;
    float acc = row[lane] * fcW[lane]
              + row[lane + 32] * fcW[lane + 32]
              + row[lane + 64] * fcW[lane + 64];
    acc = wave_sum32(acc);
    if (lane == 0) out[node] = acc + fcb[0];
}

// ================================================================ launcher

extern "C" void kernel_launch(void* const* d_in, const int* in_sizes, int n_in,
                              void* d_out, int out_size, void* d_ws, size_t ws_size,
                              hipStream_t stream) {
    const float*     x      = (const float*)d_in[0];
    const long long* eidx   = (const long long*)d_in[1];   // int64 (2, E)
    const float*     ew     = (const float*)d_in[2];
    const float*     enc_W  = (const float*)d_in[3];
    const float*     enc_b  = (const float*)d_in[4];
    const float*     Wl     = (const float*)d_in[5];
    const float*     bl     = (const float*)d_in[6];
    const float*     Wr     = (const float*)d_in[7];
    const float*     br     = (const float*)d_in[8];
    const float*     We     = (const float*)d_in[9];
    const float*     att    = (const float*)d_in[10];
    const float*     bias   = (const float*)d_in[11];
    const float*     ln_g   = (const float*)d_in[12];
    const float*     ln_b   = (const float*)d_in[13];
    const float*     lnf_g  = (const float*)d_in[14];
    const float*     lnf_b  = (const float*)d_in[15];
    const float*     fc_W   = (const float*)d_in[16];
    const float*     fc_b   = (const float*)d_in[17];
    float*           out    = (float*)d_out;

    const long long* src = eidx;
    const long long* dst = eidx + N_EDGES;

    // workspace carve-out (floats)
    float* ws = (float*)d_ws;
    float* h      = ws;                         ws += (size_t)N_NODES * HID;
    float* hn     = ws;                         ws += (size_t)N_NODES * HID;
    float* xl     = ws;                         ws += (size_t)N_NODES * HID;
    float* xr     = ws;                         ws += (size_t)N_NODES * HID;
    float* agg    = ws;                         ws += (size_t)N_NODES * HID;
    float* alpha  = ws;                         ws += (size_t)N_EDGES;
    float* aexp   = ws;                         ws += (size_t)N_EDGES;
    float* segmax = ws;                         ws += (size_t)N_NODES;
    float* segsum = ws;                         ws += (size_t)N_NODES;

    const dim3 gemm_grid(N_NODES / 16, HID / 16);   // 3125 x 6 tiles
    const dim3 wave_blk(32, 8);
    const dim3 node_grid(N_NODES / 8);              // 6250
    const dim3 edge_grid(N_EDGES / 8);              // 100000

    // encoder: h = x @ enc_W + enc_b
    k_gemm_bias<<<gemm_grid, 32, 0, stream>>>(x, enc_W, enc_b, h,
                                              N_NODES, F_IN, HID);

    for (int l = 0; l < N_LAYERS; ++l) {
        k_ln_relu<<<node_grid, wave_blk, 0, stream>>>(h, ln_g + l * HID,
                                                      ln_b + l * HID, hn);
        k_gemm_bias<<<gemm_grid, 32, 0, stream>>>(hn, Wl + (size_t)l * HID * HID,
                                                  bl + l * HID, xl,
                                                  N_NODES, HID, HID);
        k_gemm_bias<<<gemm_grid, 32, 0, stream>>>(hn, Wr + (size_t)l * HID * HID,
                                                  br + l * HID, xr,
                                                  N_NODES, HID, HID);
        k_init<<<(N_NODES * HID + 255) / 256, 256, 0, stream>>>(agg, segmax, segsum);
        k_edge_alpha<<<edge_grid, wave_blk, 0, stream>>>(xl, xr, src, dst, ew,
                                                         We + l * HID, att + l * HID,
                                                         alpha, segmax);
        k_edge_exp<<<N_EDGES / 256, 256, 0, stream>>>(alpha, segmax, dst,
                                                      aexp, segsum);
        k_edge_agg<<<edge_grid, wave_blk, 0, stream>>>(xl, src, dst, aexp,
                                                       segsum, agg);
        k_residual<<<(N_NODES * HID) / 256, 256, 0, stream>>>(h, agg, bias + l * HID);
    }

    // final LN + ReLU + FC
    k_ln_relu<<<node_grid, wave_blk, 0, stream>>>(h, lnf_g, lnf_b, hn);
    k_fc<<<node_grid, wave_blk, 0, stream>>>(hn, fc_W, fc_b, out);
}